// FastLineDarkenWithStreams_10788957847816
// MI455X (gfx1250) — compile-verified
//
#include <hip/hip_runtime.h>
#include <math.h>

#define IMG_H 2160
#define IMG_W 3840
#define HWSZ  ((size_t)IMG_H * (size_t)IMG_W)

#define TW 32                  // threads in x
#define TH 8                   // threads in y
#define VEC 4                  // pixels per thread in x
#define TILE_W (TW * VEC)      // 128 pixels per block in x
#define NTHREADS (TW * TH)     // 256 = 8 wave32

#define GS 136                 // gray-tile LDS row stride (floats, 16B-aligned rows)
#define ES 136                 // edges-tile LDS row stride
#define TS 132                 // thinned-tile LDS row stride

// ---------------------------------------------------------------------------
// CDNA5 async memory->LDS copies (ASYNCcnt path). Generic pointers to
// __shared__ carry the group-segment byte offset in their low 32 bits, which
// is exactly what the instruction's VDST (LDS address VGPR) expects.
// ---------------------------------------------------------------------------
__device__ __forceinline__ void async_copy_b32_to_lds(const float* gptr, float* lptr) {
#if defined(__HIP_DEVICE_COMPILE__) && defined(__gfx1250__)
  unsigned lds_off = (unsigned)(unsigned long long)(void*)lptr;
  asm volatile("global_load_async_to_lds_b32 %0, %1, off"
               :: "v"(lds_off), "v"(gptr) : "memory");
#else
  *lptr = *gptr;
#endif
}

__device__ __forceinline__ void async_copy_b128_to_lds(const float* gptr, float* lptr) {
#if defined(__HIP_DEVICE_COMPILE__) && defined(__gfx1250__)
  unsigned lds_off = (unsigned)(unsigned long long)(void*)lptr;
  asm volatile("global_load_async_to_lds_b128 %0, %1, off"
               :: "v"(lds_off), "v"(gptr) : "memory");
#else
  lptr[0] = gptr[0]; lptr[1] = gptr[1]; lptr[2] = gptr[2]; lptr[3] = gptr[3];
#endif
}

__device__ __forceinline__ void wait_async_lds() {
#if defined(__HIP_DEVICE_COMPILE__) && defined(__gfx1250__)
  asm volatile("s_wait_asynccnt 0" ::: "memory");
#endif
}

__device__ __forceinline__ float4 ld4(const float* p) {
  return *reinterpret_cast<const float4*>(p);
}
__device__ __forceinline__ void st4(float* p, float4 v) {
  *reinterpret_cast<float4*>(p) = v;
}

// ---------------------------------------------------------------------------
// Kernel 0: reset global min/max accumulators (ws is not re-poisoned between
// graph replays, so this must run every launch).
// ---------------------------------------------------------------------------
__global__ void init_minmax_kernel(unsigned* __restrict__ mm) {
  mm[0] = 0x7F800000u;  // +inf bits (min accumulator)
  mm[1] = 0u;           // +0.0 bits (max accumulator; edges >= 0)
}

// ---------------------------------------------------------------------------
// Kernel 1: gray -> Sobel -> |grad| -> edges buffer (b128 stores);
// global min/max reduction. Gray zero-padded outside image (conv padding=1).
// Tile: 128x8 pixels, LDS gray tile origin x = bx0-4 (16B-aligned groups).
// ---------------------------------------------------------------------------
__global__ void edges_kernel(const float* __restrict__ img,
                             float* __restrict__ edges,
                             unsigned* __restrict__ mm) {
  __shared__ __align__(16) float gT[TH + 2][GS];   // 10 x 136
  __shared__ unsigned sMin, sMax;

  const int tid = threadIdx.y * TW + threadIdx.x;
  const int bx0 = blockIdx.x * TILE_W;
  const int by0 = blockIdx.y * TH;
  const int ox  = bx0 - 4;                          // LDS x-origin (mult. of 4)

  if (tid == 0) { sMin = 0x7F800000u; sMax = 0u; }

  // Load/compute gray tile: rows gy in [by0-1, by0+8], cols gx in [ox, ox+135].
  for (int i = tid; i < (TH + 2) * (GS / 4); i += NTHREADS) {   // 10*34 = 340
    const int ly = i / (GS / 4), gq = i % (GS / 4);
    const int gy = by0 - 1 + ly;
    const int gx0 = ox + 4 * gq;
    if (gy >= 0 && gy < IMG_H && gx0 >= 0 && gx0 + 3 < IMG_W) {
      const size_t p = (size_t)gy * IMG_W + (size_t)gx0;
      const float4 r = ld4(img + p);
      const float4 g = ld4(img + p + HWSZ);
      const float4 b = ld4(img + p + 2 * HWSZ);
      float4 v;
      v.x = 0.2989f * r.x + 0.587f * g.x + 0.114f * b.x;
      v.y = 0.2989f * r.y + 0.587f * g.y + 0.114f * b.y;
      v.z = 0.2989f * r.z + 0.587f * g.z + 0.114f * b.z;
      v.w = 0.2989f * r.w + 0.587f * g.w + 0.114f * b.w;
      st4(&gT[ly][4 * gq], v);
    } else {
      for (int e = 0; e < 4; ++e) {
        const int gx = gx0 + e;
        float v = 0.0f;
        if (gy >= 0 && gy < IMG_H && gx >= 0 && gx < IMG_W) {
          const size_t p = (size_t)gy * IMG_W + (size_t)gx;
          v = 0.2989f * img[p] + 0.587f * img[p + HWSZ] + 0.114f * img[p + 2 * HWSZ];
        }
        gT[ly][4 * gq + e] = v;
      }
    }
  }
  __syncthreads();

  // Sobel + magnitude for 4 pixels per thread; vector store of edges.
  const int ty  = threadIdx.y;
  const int tx4 = threadIdx.x * VEC;
  const int gy  = by0 + ty;                 // grid exactly covers the image
  const int gx0 = bx0 + tx4;
  float ev[4];
  float vmin = __builtin_inff(), vmax = 0.0f;
#pragma unroll
  for (int j = 0; j < VEC; ++j) {
    const int c = tx4 + 4 + j;              // center col in gT (gx - ox)
    const float a00 = gT[ty    ][c - 1], a01 = gT[ty    ][c], a02 = gT[ty    ][c + 1];
    const float a10 = gT[ty + 1][c - 1],                      a12 = gT[ty + 1][c + 1];
    const float a20 = gT[ty + 2][c - 1], a21 = gT[ty + 2][c], a22 = gT[ty + 2][c + 1];
    const float ex = (a02 + 2.0f * a12 + a22) - (a00 + 2.0f * a10 + a20);
    const float ey = (a20 + 2.0f * a21 + a22) - (a00 + 2.0f * a01 + a02);
    const float e = sqrtf(ex * ex + ey * ey);
    ev[j] = e;
    vmin = fminf(vmin, e);
    vmax = fmaxf(vmax, e);
  }
  float4 e4; e4.x = ev[0]; e4.y = ev[1]; e4.z = ev[2]; e4.w = ev[3];
  st4(&edges[(size_t)gy * IMG_W + (size_t)gx0], e4);
  atomicMin(&sMin, __float_as_uint(vmin));  // e >= 0: uint order == float order
  atomicMax(&sMax, __float_as_uint(vmax));
  __syncthreads();

  if (tid == 0) {
    atomicMin(&mm[0], sMin);
    atomicMax(&mm[1], sMax);
  }
}

// ---------------------------------------------------------------------------
// Kernel 2: (normalize o erosion) -> 3x3 Gaussian -> darken + clamp.
// Erosion commutes with the positive-scale affine normalization, so we erode
// RAW edges (async b128 memory->LDS, +inf padded) and normalize once.
// Thinned tile zero-padded outside the image (conv padding=1 semantics).
// ---------------------------------------------------------------------------
__global__ void darken_kernel(const float* __restrict__ img,
                              const float* __restrict__ edges,
                              const unsigned* __restrict__ mm,
                              float* __restrict__ out) {
  __shared__ __align__(16) float eT[TH + 4][ES];   // 12 x 136 raw edges, halo 2
  __shared__ __align__(16) float tT[TH + 2][TS];   // 10 x 132 thinned, halo 1

  const int tid = threadIdx.y * TW + threadIdx.x;
  const int bx0 = blockIdx.x * TILE_W;
  const int by0 = blockIdx.y * TH;
  const int ox  = bx0 - 4;                          // eT x-origin (mult. of 4)

  // Async-stream edges tile: rows gy in [by0-2, by0+9], cols gx in [ox, ox+135].
  for (int i = tid; i < (TH + 4) * (ES / 4); i += NTHREADS) {   // 12*34 = 408
    const int ly = i / (ES / 4), gq = i % (ES / 4);
    const int gy = by0 - 2 + ly;
    const int gx0 = ox + 4 * gq;
    if (gy >= 0 && gy < IMG_H && gx0 >= 0 && gx0 + 3 < IMG_W) {
      async_copy_b128_to_lds(&edges[(size_t)gy * IMG_W + (size_t)gx0], &eT[ly][4 * gq]);
    } else {
      for (int e = 0; e < 4; ++e) {
        const int gx = gx0 + e;
        if (gy >= 0 && gy < IMG_H && gx >= 0 && gx < IMG_W) {
          async_copy_b32_to_lds(&edges[(size_t)gy * IMG_W + (size_t)gx], &eT[ly][4 * gq + e]);
        } else {
          eT[ly][4 * gq + e] = __builtin_inff();    // min identity: erosion pad
        }
      }
    }
  }
  wait_async_lds();
  __syncthreads();

  const float emin = __uint_as_float(mm[0]);
  const float emax = __uint_as_float(mm[1]);
  const float inv  = 1.0f / (emax - emin);

  // Thinned tile: positions (gy,gx) with gy in [by0-1, by0+8], gx in [bx0-1, bx0+128].
  for (int i = tid; i < (TH + 2) * (TILE_W + 2); i += NTHREADS) {  // 10*130 = 1300
    const int lr = i / (TILE_W + 2), t = i % (TILE_W + 2);
    const int gy = by0 - 1 + lr;
    const int gx = bx0 - 1 + t;
    float v = 0.0f;
    if (gy >= 0 && gy < IMG_H && gx >= 0 && gx < IMG_W) {
      const int c = t + 2;                  // eT col of (gx-1): gx-1-ox = t+2
      float m =          eT[lr    ][c];
      m = fminf(m, eT[lr    ][c + 1]); m = fminf(m, eT[lr    ][c + 2]);
      m = fminf(m, eT[lr + 1][c    ]); m = fminf(m, eT[lr + 1][c + 1]); m = fminf(m, eT[lr + 1][c + 2]);
      m = fminf(m, eT[lr + 2][c    ]); m = fminf(m, eT[lr + 2][c + 1]); m = fminf(m, eT[lr + 2][c + 2]);
      v = (m - emin) * inv;                 // normalize after erosion
    }
    tT[lr][t] = v;
  }
  __syncthreads();

  // 3x3 separable Gaussian + darken, 4 pixels per thread, b128 loads/stores.
  const int ty  = threadIdx.y;
  const int tx4 = threadIdx.x * VEC;
  const int gy  = by0 + ty;
  const int gx0 = bx0 + tx4;
  // gaussian_kernel3(sigma=1): a = e^{-1/2}/(1+2e^{-1/2}), b = 1/(1+2e^{-1/2})
  const float wv[3] = {0.27406862f, 0.45186276f, 0.27406862f};
  float dkn[4];
#pragma unroll
  for (int j = 0; j < VEC; ++j) {
    float s = 0.0f;
#pragma unroll
    for (int dy = 0; dy < 3; ++dy) {
      const int t = tx4 + j;               // tT col of (gx-1): gx-1-(bx0-1)
      const float r = wv[0] * tT[ty + dy][t] +
                      wv[1] * tT[ty + dy][t + 1] +
                      wv[2] * tT[ty + dy][t + 2];
      s += wv[dy] * r;
    }
    dkn[j] = 0.7f * s;                     // DARKEN_STRENGTH
  }

  const size_t p = (size_t)gy * IMG_W + (size_t)gx0;
#pragma unroll
  for (int ch = 0; ch < 3; ++ch) {
    const size_t q = p + (size_t)ch * HWSZ;
    const float4 c4 = ld4(img + q);
    float4 o4;
    o4.x = fminf(fmaxf(c4.x - dkn[0], 0.0f), 1.0f);
    o4.y = fminf(fmaxf(c4.y - dkn[1], 0.0f), 1.0f);
    o4.z = fminf(fmaxf(c4.z - dkn[2], 0.0f), 1.0f);
    o4.w = fminf(fmaxf(c4.w - dkn[3], 0.0f), 1.0f);
    st4(out + q, o4);
  }
}

// ---------------------------------------------------------------------------
extern "C" void kernel_launch(void* const* d_in, const int* in_sizes, int n_in,
                              void* d_out, int out_size, void* d_ws, size_t ws_size,
                              hipStream_t stream) {
  (void)in_sizes; (void)n_in; (void)out_size; (void)ws_size;

  const float* img = (const float*)d_in[0];
  float* out   = (float*)d_out;
  float* edges = (float*)d_ws;                 // HWSZ floats (~33.2 MB)
  unsigned* mm = (unsigned*)(edges + HWSZ);    // 2 uints: min/max bits

  const dim3 blk(TW, TH);
  const dim3 grd(IMG_W / TILE_W, IMG_H / TH);  // 30 x 270, exact cover

  init_minmax_kernel<<<dim3(1), dim3(1), 0, stream>>>(mm);
  edges_kernel<<<grd, blk, 0, stream>>>(img, edges, mm);
  darken_kernel<<<grd, blk, 0, stream>>>(img, edges, mm, out);
}